// EquiEncoder_87179246174224
// MI455X (gfx1250) — compile-verified
//
#include <hip/hip_runtime.h>
#include <hip/hip_bf16.h>

// ---------------------------------------------------------------------------
// EquiEncoder forward, pruned to the live dataflow for outputs (H, h).
// GEMMs: V_WMMA_F32_16X16X4_F32, one wave per 16-row strip x 128 cols
// (8 accumulators -> each A fragment feeds 8 WMMAs).
// ---------------------------------------------------------------------------

#define N_ATOM   20000
#define M_CG     2000
#define E_HALF_N 160000
#define E2       (2 * E_HALF_N)
#define F        128
#define F3       384          // row stride of Wm2/Wc2/Wdm/Wdc (we use first 128 cols)
#define NRBF     16
#define PI_F     3.14159265358979f
#define CUT_MSG_F 5.0f
#define CUT_CG_F  20.0f

typedef __attribute__((ext_vector_type(2))) float v2f;
typedef __attribute__((ext_vector_type(8))) float v8f;

__global__ void k_zero(float* __restrict__ p, int n) {
  int i = blockIdx.x * blockDim.x + threadIdx.x;
  if (i < n) p[i] = 0.0f;
}

__global__ void k_count(const int* __restrict__ mapping, float* __restrict__ cnt) {
  int i = blockIdx.x * blockDim.x + threadIdx.x;
  if (i < N_ATOM) atomicAdd(&cnt[mapping[i]], 1.0f);
}

// h[i, 0:64] = emb_atom[z[i]], h[i, 64:128] = emb_res[cg_z[mapping[i]]]
__global__ void k_init_h(const int* __restrict__ z, const int* __restrict__ cg_z,
                         const int* __restrict__ mapping,
                         const float* __restrict__ emb_atom,
                         const float* __restrict__ emb_res,
                         float* __restrict__ h) {
  int idx = blockIdx.x * blockDim.x + threadIdx.x;
  if (idx >= N_ATOM * F) return;
  int i = idx >> 7, c = idx & 127;
  float val;
  if (c < 64) val = emb_atom[z[i] * 64 + c];
  else        val = emb_res[cg_z[mapping[i]] * 64 + (c - 64)];
  h[idx] = val;
}

// C[N_ATOM x 128] = act(A[N_ATOM x 128] @ W[128 x ldw](first 128 cols) + bias)
// One wave per 16-row strip; 8 f32 accumulators covering all 128 output cols.
// K-loop: 32 steps x 8 tiles = 256 V_WMMA_F32_16X16X4_F32 per wave; each A
// fragment is reused by 8 consecutive WMMAs.
template <int ACT>
__global__ void k_gemm128(const float* __restrict__ A, const float* __restrict__ W,
                          int ldw, const float* __restrict__ bias,
                          float* __restrict__ C) {
  int wid  = (blockIdx.x * blockDim.x + threadIdx.x) >> 5;
  int lane = threadIdx.x & 31;
  if (wid >= N_ATOM / 16) return;                    // wave-uniform guard
  int row0 = wid << 4;
  int hf = lane >> 4, l = lane & 15;

  v8f acc[8];
#pragma unroll
  for (int t = 0; t < 8; ++t) acc[t] = (v8f){};

  // A frag: lane l (<16): A[row0+l][4k+0..1]; lane l+16: A[row0+l][4k+2..3]
  const float* ap = A + (size_t)(row0 + l) * F + (hf << 1);
  // B frag (mirror over N): lane l: W[4k+2*hf + {0,1}][16*t + l]
  const float* bp = W + (size_t)(hf << 1) * ldw + l;

#pragma unroll 2
  for (int k = 0; k < F / 4; ++k) {
    v2f a;
    a.x = ap[0]; a.y = ap[1];
    ap += 4;
#pragma unroll
    for (int t = 0; t < 8; ++t) {
      v2f b;
      b.x = bp[t * 16];
      b.y = bp[t * 16 + ldw];
      acc[t] = __builtin_amdgcn_wmma_f32_16x16x4_f32(
          /*neg_a=*/false, a, /*neg_b=*/false, b,
          /*c_mod=*/(short)0, acc[t], /*reuse_a=*/false, /*reuse_b=*/false);
    }
    bp += (size_t)(ldw << 2);
  }

#pragma unroll
  for (int t = 0; t < 8; ++t) {
    float bv = bias[t * 16 + l];
#pragma unroll
    for (int j = 0; j < 8; ++j) {
      float x = acc[t][j] + bv;                      // row = row0 + 8*hf + j
      if (ACT) x = x * __builtin_amdgcn_rcpf(1.0f + __expf(-x));  // silu (v_rcp)
      C[(size_t)(row0 + (hf << 3) + j) * F + t * 16 + l] = x;
    }
  }
}

// One wave per directed edge: h[src] += phi[dst] * (rbf@Wd + bd) * env  (128 ch)
__global__ void k_edge(const float* __restrict__ phi, const float* __restrict__ xyz,
                       const int* __restrict__ nbr, const float* __restrict__ Wd,
                       const float* __restrict__ bd, float* __restrict__ h) {
  int wave = (blockIdx.x * blockDim.x + threadIdx.x) >> 5;
  int lane = threadIdx.x & 31;
  if (wave >= E2) return;
  int src, dst;
  if (wave < E_HALF_N) { src = nbr[2 * wave];     dst = nbr[2 * wave + 1]; }
  else { int e = wave - E_HALF_N; src = nbr[2 * e + 1]; dst = nbr[2 * e]; }

  float rx = xyz[3 * dst + 0] - xyz[3 * src + 0];
  float ry = xyz[3 * dst + 1] - xyz[3 * src + 1];
  float rz = xyz[3 * dst + 2] - xyz[3 * src + 2];
  float d  = __fsqrt_rn(rx * rx + ry * ry + rz * rz + 3e-15f);
  float inv = __builtin_amdgcn_rcpf(d);
  float env = (d < CUT_MSG_F) ? 0.5f * (__cosf((PI_F / CUT_MSG_F) * d) + 1.0f) : 0.0f;

  float rbf[NRBF];
#pragma unroll
  for (int r = 0; r < NRBF; ++r)
    rbf[r] = __sinf((float)(r + 1) * (PI_F / CUT_MSG_F) * d) * inv;

  const float* ph = phi + (size_t)dst * F;
#pragma unroll
  for (int t = 0; t < 4; ++t) {
    int j = lane + (t << 5);
    float e0 = bd[j];
#pragma unroll
    for (int r = 0; r < NRBF; ++r)
      e0 = fmaf(rbf[r], Wd[r * F3 + j], e0);
    atomicAdd(&h[(size_t)src * F + j], ph[j] * e0 * env);
  }
}

// One wave per atom: H[mapping[i]] += h[i] / cnt   (the i==0 seeding)
__global__ void k_h_seed(const float* __restrict__ h, const int* __restrict__ mapping,
                         const float* __restrict__ cnt, float* __restrict__ H) {
  int wave = (blockIdx.x * blockDim.x + threadIdx.x) >> 5;
  int lane = threadIdx.x & 31;
  if (wave >= N_ATOM) return;
  int m = mapping[wave];
  float invc = 1.0f / fmaxf(cnt[m], 1.0f);
#pragma unroll
  for (int t = 0; t < 4; ++t) {
    int j = lane + (t << 5);
    atomicAdd(&H[(size_t)m * F + j], h[(size_t)wave * F + j] * invc);
  }
}

// One wave per atom: H[m] += phi_c[i] * (rbf(d_iI)@Wdc + bdc) * env / cnt
__global__ void k_cg(const float* __restrict__ phi, const float* __restrict__ xyz,
                     const float* __restrict__ cg_xyz, const int* __restrict__ mapping,
                     const float* __restrict__ Wd, const float* __restrict__ bd,
                     const float* __restrict__ cnt, float* __restrict__ H) {
  int wave = (blockIdx.x * blockDim.x + threadIdx.x) >> 5;
  int lane = threadIdx.x & 31;
  if (wave >= N_ATOM) return;
  int m = mapping[wave];
  float rx = xyz[3 * wave + 0] - cg_xyz[3 * m + 0];
  float ry = xyz[3 * wave + 1] - cg_xyz[3 * m + 1];
  float rz = xyz[3 * wave + 2] - cg_xyz[3 * m + 2];
  float d  = __fsqrt_rn(rx * rx + ry * ry + rz * rz + 3e-15f);
  float inv = __builtin_amdgcn_rcpf(d);
  float env = (d < CUT_CG_F) ? 0.5f * (__cosf((PI_F / CUT_CG_F) * d) + 1.0f) : 0.0f;
  float invc = 1.0f / fmaxf(cnt[m], 1.0f);

  float rbf[NRBF];
#pragma unroll
  for (int r = 0; r < NRBF; ++r)
    rbf[r] = __sinf((float)(r + 1) * (PI_F / CUT_CG_F) * d) * inv;

  const float* ph = phi + (size_t)wave * F;
#pragma unroll
  for (int t = 0; t < 4; ++t) {
    int j = lane + (t << 5);
    float e0 = bd[j];
#pragma unroll
    for (int r = 0; r < NRBF; ++r)
      e0 = fmaf(rbf[r], Wd[r * F3 + j], e0);
    atomicAdd(&H[(size_t)m * F + j], ph[j] * e0 * env * invc);
  }
}

extern "C" void kernel_launch(void* const* d_in, const int* in_sizes, int n_in,
                              void* d_out, int out_size, void* d_ws, size_t ws_size,
                              hipStream_t stream) {
  const int*   z        = (const int*)  d_in[0];
  const float* xyz      = (const float*)d_in[1];
  const int*   cg_z     = (const int*)  d_in[2];
  const float* cg_xyz   = (const float*)d_in[3];
  const int*   mapping  = (const int*)  d_in[4];
  const int*   nbr      = (const int*)  d_in[5];
  // d_in[6] cg_nbr_list, d_in[7] ic: unused by the reference's returned outputs
  const float* emb_atom = (const float*)d_in[8];
  const float* emb_res  = (const float*)d_in[9];
  const float* Wm1 = (const float*)d_in[10];
  const float* bm1 = (const float*)d_in[11];
  const float* Wm2 = (const float*)d_in[12];
  const float* bm2 = (const float*)d_in[13];
  const float* Wdm = (const float*)d_in[14];
  const float* bdm = (const float*)d_in[15];
  const float* Wc1 = (const float*)d_in[16];
  const float* bc1 = (const float*)d_in[17];
  const float* Wc2 = (const float*)d_in[18];
  const float* bc2 = (const float*)d_in[19];
  const float* Wdc = (const float*)d_in[20];
  const float* bdc = (const float*)d_in[21];
  (void)in_sizes; (void)n_in; (void)out_size; (void)ws_size;

  float* H = (float*)d_out;               // (M, 128)
  float* h = (float*)d_out + M_CG * F;    // (N, 128), accumulated in place

  float* ws   = (float*)d_ws;
  float* pre1 = ws; ws += (size_t)N_ATOM * F;   // silu(h@W1+b1)
  float* phi  = ws; ws += (size_t)N_ATOM * F;   // first 128 cols of pre1@W2+b2
  float* cnt  = ws; ws += M_CG;

  const int TB = 256;                      // 8 waves per block (wave32)
  auto blk = [](long n, long per) { return (int)((n + per - 1) / per); };

  k_zero <<<blk(M_CG,       TB), TB, 0, stream>>>(cnt, M_CG);
  k_zero <<<blk(M_CG * F,   TB), TB, 0, stream>>>(H, M_CG * F);
  k_count<<<blk(N_ATOM,     TB), TB, 0, stream>>>(mapping, cnt);
  k_init_h<<<blk((long)N_ATOM * F, TB), TB, 0, stream>>>(z, cg_z, mapping,
                                                          emb_atom, emb_res, h);

  const long gemm_waves = N_ATOM / 16;     // 1250 row strips, one wave each
  for (int L = 0; L < 2; ++L) {
    // message MLP: pre1 = silu(h@Wm1+bm1); phi = pre1@Wm2[:, :128] + bm2[:128]
    k_gemm128<1><<<blk(gemm_waves, 8), TB, 0, stream>>>(h,    Wm1 + (size_t)L * F * F,  F,
                                                        bm1 + (size_t)L * F,  pre1);
    k_gemm128<0><<<blk(gemm_waves, 8), TB, 0, stream>>>(pre1, Wm2 + (size_t)L * F * F3, F3,
                                                        bm2 + (size_t)L * F3, phi);
    // edge aggregation: h[src] += s0
    k_edge<<<blk(E2, 8), TB, 0, stream>>>(phi, xyz, nbr,
                                          Wdm + (size_t)L * NRBF * F3,
                                          bdm + (size_t)L * F3, h);
    if (L == 0)
      k_h_seed<<<blk(N_ATOM, 8), TB, 0, stream>>>(h, mapping, cnt, H);

    // CG MLP: pre1 = silu(h@Wc1+bc1); phi = pre1@Wc2[:, :128] + bc2[:128]
    k_gemm128<1><<<blk(gemm_waves, 8), TB, 0, stream>>>(h,    Wc1 + (size_t)L * F * F,  F,
                                                        bc1 + (size_t)L * F,  pre1);
    k_gemm128<0><<<blk(gemm_waves, 8), TB, 0, stream>>>(pre1, Wc2 + (size_t)L * F * F3, F3,
                                                        bc2 + (size_t)L * F3, phi);
    // CG aggregation: H[mapping] += s0 / cnt
    k_cg<<<blk(N_ATOM, 8), TB, 0, stream>>>(phi, xyz, cg_xyz, mapping,
                                            Wdc + (size_t)L * NRBF * F3,
                                            bdc + (size_t)L * F3, cnt, H);
  }
}